// StockGCN_2911987826950
// MI455X (gfx1250) — compile-verified
//
#include <hip/hip_runtime.h>
#include <hip/hip_bf16.h>

typedef __attribute__((ext_vector_type(2))) float v2f;
typedef __attribute__((ext_vector_type(8))) float v8f;

#define N_FEAT_IN 30

// ---------------------------------------------------------------- utilities

__global__ void fill_kernel(float* __restrict__ p, long long n, float v) {
    long long i = (long long)blockIdx.x * blockDim.x + threadIdx.x;
    if (i < n) p[i] = v;
}

// deg[dst[e]] += w[e]   (deg pre-filled with 1.0 == self-loop weight)
__global__ void degree_kernel(const long long* __restrict__ dst,
                              const float* __restrict__ w,
                              float* __restrict__ deg, long long E) {
    long long e = (long long)blockIdx.x * blockDim.x + threadIdx.x;
    if (e < E) unsafeAtomicAdd(&deg[dst[e]], w[e]);
}

// in-place deg -> dinv = rsqrt(max(deg,1e-12))
__global__ void dinv_kernel(float* __restrict__ deg, int N) {
    int i = blockIdx.x * blockDim.x + threadIdx.x;
    if (i < N) {
        float d = deg[i];
        deg[i] = (d > 0.0f) ? rsqrtf(fmaxf(d, 1e-12f)) : 0.0f;
    }
}

// norm[e] = dinv[src] * w * dinv[dst]
__global__ void norm_kernel(const long long* __restrict__ src,
                            const long long* __restrict__ dst,
                            const float* __restrict__ w,
                            const float* __restrict__ dinv,
                            float* __restrict__ norm, long long E) {
    long long e = (long long)blockIdx.x * blockDim.x + threadIdx.x;
    if (e < E) norm[e] = dinv[src[e]] * w[e] * dinv[dst[e]];
}

// pad W1 [30 x 64] row-major into Wp [32 x 64], rows 30..31 = 0
__global__ void padW_kernel(const float* __restrict__ W, float* __restrict__ Wp) {
    int i = blockIdx.x * blockDim.x + threadIdx.x;   // 0 .. 32*64-1
    if (i < 32 * 64) Wp[i] = (i < N_FEAT_IN * 64) ? W[i] : 0.0f;
}

// ---------------------------------------------------------------- aggregation
// out[dst, f..f+VEC) += feat[src, f..f+VEC) * norm  over E edges + N self loops.
// blockIdx.x covers edges, blockIdx.y covers feature chunks (f = blockIdx.y*VEC).
// Self loop e' = e - E: src = dst = e', norm = dinv[e']^2.
template <int VEC>
__global__ __launch_bounds__(256)
void aggregate_kernel(const float* __restrict__ feat, int ldf,
                      const long long* __restrict__ src,
                      const long long* __restrict__ dst,
                      const float* __restrict__ norm,
                      const float* __restrict__ dinv,
                      float* __restrict__ out, int ldo,
                      long long E, int N) {
    long long e = (long long)blockIdx.x * blockDim.x + threadIdx.x;
    if (e >= E + (long long)N) return;
    int f = blockIdx.y * VEC;

    long long s, d;
    float nv;
    if (e < E) {
        s = src[e];
        d = dst[e];
        nv = norm[e];
    } else {
        s = d = e - E;
        float di = dinv[s];
        nv = di * di;
    }

    const float* fp = feat + s * ldf + f;
    float* op = out + d * ldo + f;
    float v[VEC];
#pragma unroll
    for (int i = 0; i < VEC; ++i) v[i] = fp[i] * nv;   // merges into b64/b128 load
#pragma unroll
    for (int i = 0; i < VEC; ++i) unsafeAtomicAdd(&op[i], v[i]);
}

// ---------------------------------------------------------------- WMMA GEMM
// out = relu(in[N x KPAD] @ W[KPAD x 64] + bias), fp32 WMMA 16x16x4.
// in is zero-padded to KPAD columns; W is zero-padded to KPAD rows -> no guards.
// One wave computes one 16x16 output tile; 8 waves per block.
//
// A frag (16x4 f32, ISA 7.12.2): lanes 0-15 hold M=lane, VGPR0=K0,VGPR1=K1;
//                                lanes 16-31 hold M=lane-16, VGPR0=K2,VGPR1=K3.
// B frag (4x16): row-striped across lanes: VGPR0 = {K0 | K2}, VGPR1 = {K1 | K3}.
// C/D (16x16 f32): VGPR r -> M = r + 8*(lane>>4), N = lane&15.
template <int KPAD>
__global__ __launch_bounds__(256)
void gemm_wmma_kernel(const float* __restrict__ in,
                      const float* __restrict__ W,
                      const float* __restrict__ bias,
                      float* __restrict__ out,
                      int nTilesN, int totalTiles) {
    int wave = (blockIdx.x << 3) + (threadIdx.x >> 5);
    if (wave >= totalTiles) return;                    // uniform per wave
    int lane = threadIdx.x & 31;
    int mTile = wave / nTilesN;
    int nTile = wave - mTile * nTilesN;

    int halfsel = lane >> 4;                           // 0: lanes 0-15, 1: 16-31
    int lanelo  = lane & 15;
    const float* arow = in + ((long long)mTile * 16 + lanelo) * KPAD + halfsel * 2;
    int col = nTile * 16 + lanelo;

    v8f c = {};
#pragma unroll
    for (int k = 0; k < KPAD; k += 4) {
        v2f a = *(const v2f*)(arow + k);               // A: one aligned b64 load
        int kk = k + halfsel * 2;
        v2f b;
        b[0] = W[kk * 64 + col];                       // B: two cached b32 loads
        b[1] = W[(kk + 1) * 64 + col];
        c = __builtin_amdgcn_wmma_f32_16x16x4_f32(
                false, a, false, b, (short)0, c, false, false);
    }

    float bv = bias[col];
    float* orow = out + ((long long)mTile * 16 + halfsel * 8) * 64 + col;
#pragma unroll
    for (int r = 0; r < 8; ++r)
        orow[r * 64] = fmaxf(c[r] + bv, 0.0f);         // fused bias + ReLU
}

// ---------------------------------------------------------------- layer 3 GEMV
// t[i] = H2[i,:] (64) . W3 (64x1)
__global__ void gemv64_kernel(const float* __restrict__ h,
                              const float* __restrict__ W3,
                              float* __restrict__ t, int N) {
    int i = blockIdx.x * blockDim.x + threadIdx.x;
    if (i >= N) return;
    const float* r = h + (long long)i * 64;
    float acc = 0.0f;
#pragma unroll
    for (int k = 0; k < 64; ++k) acc += r[k] * W3[k];
    t[i] = acc;
}

// out[i] = relu(out[i] + b3)
__global__ void finalize_kernel(float* __restrict__ out,
                                const float* __restrict__ b3, int N) {
    int i = blockIdx.x * blockDim.x + threadIdx.x;
    if (i < N) out[i] = fmaxf(out[i] + b3[0], 0.0f);
}

// ---------------------------------------------------------------- launcher

extern "C" void kernel_launch(void* const* d_in, const int* in_sizes, int n_in,
                              void* d_out, int out_size, void* d_ws, size_t ws_size,
                              hipStream_t stream) {
    const float*     x  = (const float*)d_in[0];
    const long long* ei = (const long long*)d_in[1];   // [2, E] int64
    const float*     ew = (const float*)d_in[2];
    const float*     W1 = (const float*)d_in[3];
    const float*     b1 = (const float*)d_in[4];
    const float*     W2 = (const float*)d_in[5];
    const float*     b2 = (const float*)d_in[6];
    const float*     W3 = (const float*)d_in[7];
    const float*     b3 = (const float*)d_in[8];

    const int       N = in_sizes[0] / N_FEAT_IN;       // 100000
    const long long E = in_sizes[2];                   // 3200000
    const long long* src = ei;
    const long long* dst = ei + E;

    // workspace carve-up (floats)
    float* ws   = (float*)d_ws;
    float* dinv = ws;                                  // N   (deg -> dinv)
    float* norm = dinv + N;                            // E
    float* P0   = norm + E;                            // N*64
    float* P1   = P0 + (size_t)N * 64;                 // N*64
    float* T3   = P1 + (size_t)N * 64;                 // N
    float* W1p  = T3 + N;                              // 32*64 padded W1
    float* out  = (float*)d_out;                       // N

    const int B = 256;
    auto blocks = [](long long n, int b) { return (unsigned)((n + b - 1) / b); };
    const unsigned edgeBlocks = blocks(E + N, B);

    // 1. degree (deg init = 1.0 covers self-loop weight), dinv, per-edge norm
    fill_kernel<<<blocks(N, B), B, 0, stream>>>(dinv, N, 1.0f);
    degree_kernel<<<blocks(E, B), B, 0, stream>>>(dst, ew, dinv, E);
    dinv_kernel<<<blocks(N, B), B, 0, stream>>>(dinv, N);
    norm_kernel<<<blocks(E, B), B, 0, stream>>>(src, dst, ew, dinv, norm, E);
    padW_kernel<<<8, 256, 0, stream>>>(W1, W1p);

    // 2. layer 1: AX = aggregate(X) (30-wide, stride-32 padded), H1 = relu(AX@W1+b1)
    fill_kernel<<<blocks((long long)N * 32, B), B, 0, stream>>>(P0, (long long)N * 32, 0.0f);
    {
        dim3 g(edgeBlocks, N_FEAT_IN / 2);             // VEC=2, 15 chunks
        aggregate_kernel<2><<<g, B, 0, stream>>>(x, N_FEAT_IN, src, dst, norm, dinv,
                                                 P0, 32, E, N);
    }
    {
        int nTilesN = 4, totalTiles = (N / 16) * nTilesN;
        gemm_wmma_kernel<32><<<(totalTiles + 7) / 8, 256, 0, stream>>>(
            P0, W1p, b1, P1, nTilesN, totalTiles);
    }

    // 3. layer 2: AH1 = aggregate(H1), H2 = relu(AH1@W2+b2)
    fill_kernel<<<blocks((long long)N * 64, B), B, 0, stream>>>(P0, (long long)N * 64, 0.0f);
    {
        dim3 g(edgeBlocks, 64 / 4);                    // VEC=4, 16 chunks
        aggregate_kernel<4><<<g, B, 0, stream>>>(P1, 64, src, dst, norm, dinv,
                                                 P0, 64, E, N);
    }
    {
        int nTilesN = 4, totalTiles = (N / 16) * nTilesN;
        gemm_wmma_kernel<64><<<(totalTiles + 7) / 8, 256, 0, stream>>>(
            P0, W2, b2, P1, nTilesN, totalTiles);
    }

    // 4. layer 3: GEMV first (64 -> 1), then 1-wide aggregation, bias+relu
    gemv64_kernel<<<blocks(N, B), B, 0, stream>>>(P1, W3, T3, N);
    fill_kernel<<<blocks(N, B), B, 0, stream>>>(out, N, 0.0f);
    {
        dim3 g(edgeBlocks, 1);                         // VEC=1
        aggregate_kernel<1><<<g, B, 0, stream>>>(T3, 1, src, dst, norm, dinv,
                                                 out, 1, E, N);
    }
    finalize_kernel<<<blocks(N, B), B, 0, stream>>>(out, b3, N);
}